// GCNEncoder_38654705664006
// MI455X (gfx1250) — compile-verified
//
#include <hip/hip_runtime.h>

typedef __attribute__((ext_vector_type(2))) float v2f;
typedef __attribute__((ext_vector_type(8))) float v8f;

#define DFEAT 64

// ---------------------------------------------------------------------------
// Degree / normalization kernels
// ---------------------------------------------------------------------------
__global__ void k_fill1(float* __restrict__ deg, int n) {
    int i = blockIdx.x * blockDim.x + threadIdx.x;
    if (i < n) deg[i] = 1.0f;  // self loop contributes 1
}

__global__ void k_deg(const long long* __restrict__ dst, float* __restrict__ deg, int E) {
    int e = blockIdx.x * blockDim.x + threadIdx.x;
    if (e < E) atomicAdd(&deg[(int)dst[e]], 1.0f);
}

__global__ void k_dinv(const float* __restrict__ deg, float* __restrict__ dinv, int n) {
    int i = blockIdx.x * blockDim.x + threadIdx.x;
    if (i < n) dinv[i] = rsqrtf(deg[i]);  // deg >= 1 always
}

// ---------------------------------------------------------------------------
// LDS layout for B fragments: W[k][n] and W[k+1][n] stored adjacent so one
// ds_load_b64 yields an even-aligned WMMA-ready v2f. Odd K-pairs offset by
// 32 floats so khalf=0 (pair 2ks) and khalf=1 (pair 2ks+1) half-waves hit
// disjoint halves of the 64 LDS banks (no 2-way conflict on b64 reads).
// ---------------------------------------------------------------------------
__device__ __forceinline__ int wofs(int k, int n) {
    int p = k >> 1;
    return p * 128 + (p & 1) * 32 + n * 2 + (k & 1);
}
#define SW2_SIZE (32 * 128 + 64)

// ---------------------------------------------------------------------------
// WMMA GEMM: out[N x 64] = act(in)[N x 64] @ W[64 x 64]
// fp32 WMMA 16x16x4. One wave per 16-row strip, 4 col tiles, K=64 in 16 steps.
// ---------------------------------------------------------------------------
template <int RELU>
__global__ __launch_bounds__(256) void k_gemm_wmma(const float* __restrict__ in,
                                                   const float* __restrict__ W,
                                                   float* __restrict__ out,
                                                   int nrows) {
    __shared__ __align__(16) float sW2[SW2_SIZE];
    int t = threadIdx.x;
#pragma unroll
    for (int i = 0; i < 16; ++i) {
        int g = t + i * 256;          // flat index into W
        int k = g >> 6;
        int n = g & 63;
        sW2[wofs(k, n)] = W[g];
    }
    __syncthreads();

    int wid  = t >> 5;
    int lane = t & 31;
    int mtile = blockIdx.x * 8 + wid;            // wave-uniform
    if (mtile * 16 >= nrows) return;

    int mrow  = lane & 15;       // M (or N for B/C) within tile
    int khalf = lane >> 4;       // 0: K+{0,1}, 1: K+{2,3}
    const float* arow = in + (size_t)(mtile * 16 + mrow) * DFEAT;

    v8f c0 = {}, c1 = {}, c2 = {}, c3 = {};
    int nc = mrow;

#pragma unroll
    for (int ks = 0; ks < 16; ++ks) {
        int k0 = ks * 4 + khalf * 2;

        // A fragment: two consecutive K values of this lane's row
        float a0 = arow[k0];
        float a1 = arow[k0 + 1];
        if (RELU) { a0 = fmaxf(a0, 0.0f); a1 = fmaxf(a1, 0.0f); }
        v2f a; a.x = a0; a.y = a1;

        // B fragments: single b64 loads, already WMMA-laid-out
        v2f b0 = *(const v2f*)&sW2[wofs(k0, nc)];
        v2f b1 = *(const v2f*)&sW2[wofs(k0, nc + 16)];
        v2f b2 = *(const v2f*)&sW2[wofs(k0, nc + 32)];
        v2f b3 = *(const v2f*)&sW2[wofs(k0, nc + 48)];

        c0 = __builtin_amdgcn_wmma_f32_16x16x4_f32(false, a, false, b0, (short)0, c0, false, false);
        c1 = __builtin_amdgcn_wmma_f32_16x16x4_f32(false, a, false, b1, (short)0, c1, false, false);
        c2 = __builtin_amdgcn_wmma_f32_16x16x4_f32(false, a, false, b2, (short)0, c2, false, false);
        c3 = __builtin_amdgcn_wmma_f32_16x16x4_f32(false, a, false, b3, (short)0, c3, false, false);
    }

    // C layout: VGPR v -> row v + 8*(lane>>4), col lane&15
    float* orow = out + (size_t)mtile * 16 * DFEAT;
    int rbase = khalf * 8;
#pragma unroll
    for (int v = 0; v < 8; ++v) {
        int r = rbase + v;
        orow[r * DFEAT + nc]      = c0[v];
        orow[r * DFEAT + 16 + nc] = c1[v];
        orow[r * DFEAT + 32 + nc] = c2[v];
        orow[r * DFEAT + 48 + nc] = c3[v];
    }
}

// ---------------------------------------------------------------------------
// out[i][f] = b[f] + dinv[i]^2 * hw[i][f]   (bias + self-loop folded)
// One float4 per thread.
// ---------------------------------------------------------------------------
__global__ void k_init_out(const float* __restrict__ hw, const float* __restrict__ dinv,
                           const float* __restrict__ b, float* __restrict__ out, int n) {
    int i = blockIdx.x * blockDim.x + threadIdx.x;  // over n*16 float4s
    if (i < n * (DFEAT / 4)) {
        int row = i >> 4;
        int f4  = (i & 15) * 4;
        float di = dinv[row];
        float d2 = di * di;
        float4 h = *(const float4*)&hw[(size_t)row * DFEAT + f4];
        float4 bb = *(const float4*)&b[f4];
        float4 o;
        o.x = bb.x + d2 * h.x;
        o.y = bb.y + d2 * h.y;
        o.z = bb.z + d2 * h.z;
        o.w = bb.w + d2 * h.w;
        *(float4*)&out[(size_t)row * DFEAT + f4] = o;
    }
}

// ---------------------------------------------------------------------------
// Edge scatter: out[dst] += dinv[src]*dinv[dst] * hw[src]
// One (edge, feature-pair) per thread: float2 gather (L2 hit), 2 f32 atomics.
// ---------------------------------------------------------------------------
__global__ void k_scatter(const long long* __restrict__ src, const long long* __restrict__ dst,
                          const float* __restrict__ hw, const float* __restrict__ dinv,
                          float* __restrict__ out, int E) {
    long long gid = (long long)blockIdx.x * blockDim.x + threadIdx.x;
    if (gid >= (long long)E * (DFEAT / 2)) return;
    int e  = (int)(gid >> 5);
    int f2 = (int)(gid & 31) * 2;
    int s = (int)src[e];
    int d = (int)dst[e];
    float nrm = dinv[s] * dinv[d];
    float2 h = *(const float2*)&hw[s * DFEAT + f2];
    atomicAdd(&out[d * DFEAT + f2],     nrm * h.x);
    atomicAdd(&out[d * DFEAT + f2 + 1], nrm * h.y);
}

// ---------------------------------------------------------------------------
extern "C" void kernel_launch(void* const* d_in, const int* in_sizes, int n_in,
                              void* d_out, int out_size, void* d_ws, size_t ws_size,
                              hipStream_t stream) {
    const float*     x    = (const float*)d_in[0];
    const long long* eidx = (const long long*)d_in[1];
    const float* W1 = (const float*)d_in[2];
    const float* b1 = (const float*)d_in[3];
    const float* W2 = (const float*)d_in[4];
    const float* b2 = (const float*)d_in[5];
    const float* W3 = (const float*)d_in[6];
    const float* b3 = (const float*)d_in[7];

    int N = in_sizes[0] / DFEAT;
    int E = in_sizes[1] / 2;
    const long long* src = eidx;
    const long long* dst = eidx + E;

    // Workspace layout
    char* ws = (char*)d_ws;
    float* dinv = (float*)ws;  ws += ((size_t)N * 4 + 255) & ~(size_t)255;
    float* hw   = (float*)ws;  ws += (size_t)N * DFEAT * 4;
    float* h1   = (float*)ws;  ws += (size_t)N * DFEAT * 4;
    float* h2   = (float*)ws;
    float* deg  = hw;  // deg is dead once dinv exists; alias onto hw

    const int TB = 256;
    int nb_N  = (N + TB - 1) / TB;
    int nb_E  = (E + TB - 1) / TB;
    int nb_I  = (N * (DFEAT / 4) + TB - 1) / TB;
    int nb_S  = (int)(((long long)E * (DFEAT / 2) + TB - 1) / TB);
    int nb_G  = (N / 16 + 7) / 8;  // 8 waves (M-tiles) per block

    // Normalization
    k_fill1<<<nb_N, TB, 0, stream>>>(deg, N);
    k_deg  <<<nb_E, TB, 0, stream>>>(dst, deg, E);
    k_dinv <<<nb_N, TB, 0, stream>>>(deg, dinv, N);

    float* outp = (float*)d_out;

    // Layer 1: hw = x @ W1 ; h1 = b1 + norm-aggregated hw
    k_gemm_wmma<0><<<nb_G, TB, 0, stream>>>(x, W1, hw, N);
    k_init_out<<<nb_I, TB, 0, stream>>>(hw, dinv, b1, h1, N);
    k_scatter <<<nb_S, TB, 0, stream>>>(src, dst, hw, dinv, h1, E);

    // Layer 2: relu(h1) applied at GEMM A-load
    k_gemm_wmma<1><<<nb_G, TB, 0, stream>>>(h1, W2, hw, N);
    k_init_out<<<nb_I, TB, 0, stream>>>(hw, dinv, b2, h2, N);
    k_scatter <<<nb_S, TB, 0, stream>>>(src, dst, hw, dinv, h2, E);

    // Layer 3: relu(h2) at A-load, no relu on output
    k_gemm_wmma<1><<<nb_G, TB, 0, stream>>>(h2, W3, hw, N);
    k_init_out<<<nb_I, TB, 0, stream>>>(hw, dinv, b3, outp, N);
    k_scatter <<<nb_S, TB, 0, stream>>>(src, dst, hw, dinv, outp, E);
}